// GCN_18433999635132
// MI455X (gfx1250) — compile-verified
//
#include <hip/hip_runtime.h>
#include <hip/hip_bf16.h>
#include <math.h>

// ---------------------------------------------------------------------------
// GCN + LSTM actor-critic forward, fp32, WMMA 16x16x4 F32 tiles (CDNA5 wave32)
// Round 2: 2x2 register-blocked GEMM (1 b64 load per WMMA), fused one-kernel
// LSTM step (WMMA -> LDS -> gate math) with ping-pong h buffers.
// ---------------------------------------------------------------------------
typedef __attribute__((ext_vector_type(2))) float v2f;
typedef __attribute__((ext_vector_type(8))) float v8f;

#define BSZ   16
#define TT    32
#define BB    512      // BSZ*TT
#define NOBJ  100
#define NPAD  112      // 7*16
#define GLOVE 300
#define GEMB  64
#define ADIM  10
#define GCND  512
#define HD    512
#define ASP   6
#define RESD  512
#define FIN   1098     // RESD+GEMB+ADIM+GCND
#define FPAD  1104     // next multiple of 16
#define CHUNK 128

static inline int cdiv(int a, int b) { return (a + b - 1) / b; }

__device__ __forceinline__ v8f wmma4(v8f acc, v2f a, v2f b) {
  // D = A(16x4 f32) x B(4x16 f32) + C(16x16 f32)
  return __builtin_amdgcn_wmma_f32_16x16x4_f32(false, a, false, b, (short)0,
                                               acc, false, false);
}

__device__ __forceinline__ float sigm(float x) { return 1.0f / (1.0f + expf(-x)); }

__device__ __forceinline__ void store_tile(float* __restrict__ C, long ldc,
                                           long mB, long nB, v8f acc, float bv,
                                           int doRelu, int half, int l16) {
#pragma unroll
  for (int i = 0; i < 8; ++i) {
    float v = acc[i] + bv;
    if (doRelu) v = fmaxf(v, 0.0f);
    C[(mB + i + 8 * half) * ldc + nB + l16] = v;
  }
}

// C[M,N] = act(X[M,K] @ W[N,K]^T + bias), 16x16/wave. grid=(N/16, M/16).
__global__ __launch_bounds__(32) void gemm_xwt(
    const float* __restrict__ X, const float* __restrict__ W,
    const float* __restrict__ bias, float* __restrict__ C,
    int K, int ldx, int ldw, int ldc, int doRelu) {
  const int lane = threadIdx.x & 31;
  const int half = lane >> 4;
  const int l16  = lane & 15;
  const long mBase = (long)blockIdx.y * 16;
  const long nBase = (long)blockIdx.x * 16;
  const float* xr = X + (mBase + l16) * (long)ldx;
  const float* wr = W + (nBase + l16) * (long)ldw;
  v8f acc = {};
  for (int k0 = 0; k0 < K; k0 += 4) {
    const int ka = k0 + (half << 1);
    v2f a, b;
    a.x = xr[ka]; a.y = xr[ka + 1];
    b.x = wr[ka]; b.y = wr[ka + 1];
    acc = wmma4(acc, a, b);
  }
  const float bv = bias ? bias[nBase + l16] : 0.0f;
  store_tile(C, ldc, mBase, nBase, acc, bv, doRelu, half, l16);
}

// 2x2 register-blocked: 32x32 of C per wave -> 1 b64 load per WMMA.
// grid=(N/32, M/32). Requires M%32==0, N%32==0.
__global__ __launch_bounds__(32) void gemm_xwt2(
    const float* __restrict__ X, const float* __restrict__ W,
    const float* __restrict__ bias, float* __restrict__ C,
    int K, int ldx, int ldw, int ldc, int doRelu) {
  const int lane = threadIdx.x & 31;
  const int half = lane >> 4;
  const int l16  = lane & 15;
  const long mBase = (long)blockIdx.y * 32;
  const long nBase = (long)blockIdx.x * 32;
  const float* xr0 = X + (mBase + l16) * (long)ldx;
  const float* xr1 = xr0 + 16 * (long)ldx;
  const float* wr0 = W + (nBase + l16) * (long)ldw;
  const float* wr1 = wr0 + 16 * (long)ldw;
  v8f acc00 = {}, acc01 = {}, acc10 = {}, acc11 = {};
  for (int k0 = 0; k0 < K; k0 += 4) {
    const int ka = k0 + (half << 1);
    v2f a0, a1, b0, b1;
    a0.x = xr0[ka]; a0.y = xr0[ka + 1];
    a1.x = xr1[ka]; a1.y = xr1[ka + 1];
    b0.x = wr0[ka]; b0.y = wr0[ka + 1];
    b1.x = wr1[ka]; b1.y = wr1[ka + 1];
    acc00 = wmma4(acc00, a0, b0);
    acc01 = wmma4(acc01, a0, b1);
    acc10 = wmma4(acc10, a1, b0);
    acc11 = wmma4(acc11, a1, b1);
  }
  const float bv0 = bias ? bias[nBase + l16] : 0.0f;
  const float bv1 = bias ? bias[nBase + 16 + l16] : 0.0f;
  store_tile(C, ldc, mBase,      nBase,      acc00, bv0, doRelu, half, l16);
  store_tile(C, ldc, mBase,      nBase + 16, acc01, bv1, doRelu, half, l16);
  store_tile(C, ldc, mBase + 16, nBase,      acc10, bv0, doRelu, half, l16);
  store_tile(C, ldc, mBase + 16, nBase + 16, acc11, bv1, doRelu, half, l16);
}

// C[z][M,N] = X[M,K] @ Y[z][K,N] (X shared). grid=(N/16, M/16, batch).
__global__ __launch_bounds__(32) void gemm_xy_b(
    const float* __restrict__ X, const float* __restrict__ Y,
    float* __restrict__ C, int K, int ldx, int ldy, int ldc,
    long yStride, long cStride) {
  const int lane = threadIdx.x & 31;
  const int half = lane >> 4;
  const int l16  = lane & 15;
  const long mBase = (long)blockIdx.y * 16;
  const long nBase = (long)blockIdx.x * 16;
  const float* Yb = Y + (long)blockIdx.z * yStride;
  float* Cb = C + (long)blockIdx.z * cStride;
  const float* xr = X + (mBase + l16) * (long)ldx;
  v8f acc = {};
  for (int k0 = 0; k0 < K; k0 += 4) {
    const int ka = k0 + (half << 1);
    v2f a, b;
    a.x = xr[ka]; a.y = xr[ka + 1];
    b.x = Yb[(long)ka * ldy + nBase + l16];
    b.y = Yb[(long)(ka + 1) * ldy + nBase + l16];
    acc = wmma4(acc, a, b);
  }
#pragma unroll
  for (int i = 0; i < 8; ++i)
    Cb[(mBase + i + 8 * half) * (long)ldc + nBase + l16] = acc[i];
}

// One fused LSTM timestep. grid = 32 blocks (j slices of 16), block = 128 (4
// waves). Wave w computes gate-w WMMA tile g[:, w*512 + jslice] = h @ Whh^T
// into LDS; after barrier the block applies gate math for its 16 columns.
// h is ping-pong buffered across steps (hsrc read-only here, hdst written).
__global__ __launch_bounds__(128) void k_lstm_step(
    const float* __restrict__ Whh, const float* __restrict__ Xih,
    const float* __restrict__ hsrc, float* __restrict__ hdst,
    float* __restrict__ cbuf, float* __restrict__ sf, int t) {
  __shared__ float gtile[4][16][16];   // [gate][m][j]
  const int wave = threadIdx.x >> 5;   // 0..3 == gate index (i,f,g,o)
  const int lane = threadIdx.x & 31;
  const int half = lane >> 4;
  const int l16  = lane & 15;
  const int jBase = blockIdx.x * 16;
  const long nCol = (long)wave * HD + jBase;     // column base in [0,2048)
  const float* xr = hsrc + (long)l16 * HD;       // M = 16 batch rows
  const float* wr = Whh + (nCol + l16) * (long)HD;
  v8f acc = {};
  for (int k0 = 0; k0 < HD; k0 += 4) {
    const int ka = k0 + (half << 1);
    v2f a, b;
    a.x = xr[ka]; a.y = xr[ka + 1];
    b.x = wr[ka]; b.y = wr[ka + 1];
    acc = wmma4(acc, a, b);
  }
#pragma unroll
  for (int i = 0; i < 8; ++i)
    gtile[wave][i + 8 * half][l16] = acc[i];
  __syncthreads();
  // 256 gate elements, 128 threads -> 2 each
  for (int e = threadIdx.x; e < BSZ * 16; e += 128) {
    const int b  = e >> 4;
    const int jj = e & 15;
    const int j  = jBase + jj;
    const float* xih = Xih + (long)(b * TT + t) * (4 * HD);
    const float gi = gtile[0][b][jj] + xih[j];
    const float gf = gtile[1][b][jj] + xih[HD + j];
    const float gg = gtile[2][b][jj] + xih[2 * HD + j];
    const float go = gtile[3][b][jj] + xih[3 * HD + j];
    const int idx = b * HD + j;
    const float c = sigm(gf) * cbuf[idx] + sigm(gi) * tanhf(gg);
    const float h = sigm(go) * tanhf(c);
    cbuf[idx] = c;
    hdst[idx] = h;
    sf[(long)(b * TT + t) * HD + j] = h;
  }
}

// ---------------- elementwise / small kernels ----------------
__global__ void k_pack_apad(const float* __restrict__ A, float* __restrict__ Apad) {
  int i = blockIdx.x * blockDim.x + threadIdx.x;
  if (i >= NPAD * NPAD) return;
  int n = i / NPAD, m = i % NPAD;
  Apad[i] = (n < NOBJ && m < NOBJ) ? A[n * NOBJ + m] : 0.0f;
}
__global__ void k_rowsum(const float* __restrict__ Apad, float* __restrict__ r) {
  int n = blockIdx.x * blockDim.x + threadIdx.x;
  if (n >= NPAD) return;
  float s = 0.0f;
  for (int m = 0; m < NPAD; ++m) s += Apad[n * NPAD + m];
  r[n] = s;
}
__global__ void k_word(const float* __restrict__ G, const float* __restrict__ Ww,
                       const float* __restrict__ bw, float* __restrict__ wordP) {
  int i = blockIdx.x * blockDim.x + threadIdx.x;
  if (i >= NPAD * GEMB) return;
  int n = i / GEMB, j = i % GEMB;
  if (n >= NOBJ) { wordP[i] = 0.0f; return; }
  float s = bw[j];
  for (int k = 0; k < GLOVE; ++k) s += G[n * GLOVE + k] * Ww[j * GLOVE + k];
  wordP[i] = s;                                       // no relu (reference)
}
__global__ void k_acte(const float* __restrict__ ap, const float* __restrict__ Wa,
                       const float* __restrict__ ba, float* __restrict__ ae) {
  int i = blockIdx.x * blockDim.x + threadIdx.x;
  if (i >= BB * ADIM) return;
  int b = i / ADIM, j = i % ADIM;
  float s = ba[j];
  for (int k = 0; k < ASP; ++k) s += ap[b * ASP + k] * Wa[j * ASP + k];
  ae[i] = fmaxf(s, 0.0f);
}
// x1[bb,n,d] = relu(r[n]*F1[b0+bb,d] + G1[n,d]); pad rows (n>=100) come out 0.
__global__ void k_x1(const float* __restrict__ F1, const float* __restrict__ G1,
                     const float* __restrict__ r, float* __restrict__ x1, int b0) {
  int i = blockIdx.x * blockDim.x + threadIdx.x;
  if (i >= CHUNK * NPAD * 256) return;
  int d = i % 256;
  int n = (i / 256) % NPAD;
  int bb = i / (256 * NPAD);
  float v = r[n] * F1[(long)(b0 + bb) * 256 + d] + G1[n * 256 + d];
  x1[i] = fmaxf(v, 0.0f);
}
__global__ void k_vdot(const float* __restrict__ x2, const float* __restrict__ W2,
                       float* __restrict__ v, int b0) {
  int row = blockIdx.x * blockDim.x + threadIdx.x;
  if (row >= CHUNK * NPAD) return;
  const float* xr = x2 + (long)row * 256;
  float s = 0.0f;
  for (int d = 0; d < 256; ++d) s += xr[d] * W2[d];
  v[(long)b0 * NPAD + row] = s;
}
__global__ void k_wfmp(const float* __restrict__ Wfm, float* __restrict__ WfmP) {
  int i = blockIdx.x * blockDim.x + threadIdx.x;
  if (i >= GCND * NPAD) return;
  int j = i / NPAD, m = i % NPAD;
  WfmP[i] = (m < NOBJ) ? Wfm[j * NOBJ + m] : 0.0f;
}
__global__ void k_fused(const float* __restrict__ feats, const float* __restrict__ ge,
                        const float* __restrict__ ae, const float* __restrict__ gc,
                        float* __restrict__ fp) {
  int i = blockIdx.x * blockDim.x + threadIdx.x;
  if (i >= BB * FPAD) return;
  int b = i / FPAD, c = i % FPAD;
  float v;
  if (c < RESD)                    v = feats[(long)b * RESD + c];
  else if (c < RESD + GEMB)        v = ge[b * GEMB + (c - RESD)];
  else if (c < RESD + GEMB + ADIM) v = ae[b * ADIM + (c - RESD - GEMB)];
  else if (c < FIN)                v = gc[(long)b * GCND + (c - RESD - GEMB - ADIM)];
  else                             v = 0.0f;
  fp[i] = v;
}
__global__ void k_wfp(const float* __restrict__ Wf, float* __restrict__ WfP) {
  int i = blockIdx.x * blockDim.x + threadIdx.x;
  if (i >= HD * FPAD) return;
  int j = i / FPAD, c = i % FPAD;
  WfP[i] = (c < FIN) ? Wf[(long)j * FIN + c] : 0.0f;
}
__global__ void k_bsum(const float* __restrict__ a, const float* __restrict__ b,
                       float* __restrict__ o) {
  int i = blockIdx.x * blockDim.x + threadIdx.x;
  if (i < 4 * HD) o[i] = a[i] + b[i];
}
__global__ void k_zero(float* __restrict__ p, int n) {
  int i = blockIdx.x * blockDim.x + threadIdx.x;
  if (i < n) p[i] = 0.0f;
}
__global__ void k_actor(const float* __restrict__ a1, const float* __restrict__ Wa2,
                        const float* __restrict__ ba2, float* __restrict__ out) {
  int i = blockIdx.x * blockDim.x + threadIdx.x;
  if (i >= BB * ASP) return;
  int b = i / ASP, a = i % ASP;
  float s = ba2[a];
  for (int k = 0; k < 128; ++k) s += a1[b * 128 + k] * Wa2[a * 128 + k];
  out[i] = fmaxf(s, 0.0f);
}
__global__ void k_critic(const float* __restrict__ c1, const float* __restrict__ Wc2,
                         const float* __restrict__ bc2, float* __restrict__ out) {
  int b = blockIdx.x * blockDim.x + threadIdx.x;
  if (b >= BB) return;
  float s = bc2[0];
  for (int k = 0; k < 128; ++k) s += c1[b * 128 + k] * Wc2[k];
  out[b] = fmaxf(s, 0.0f);
}
__global__ void k_hcout(const float* __restrict__ h, const float* __restrict__ c,
                        float* __restrict__ oh, float* __restrict__ oc) {
  int i = blockIdx.x * blockDim.x + threadIdx.x;
  if (i >= BSZ * HD) return;
  oh[i] = h[i];
  oc[i] = c[i];
}

// ---------------------------------------------------------------------------
extern "C" void kernel_launch(void* const* d_in, const int* in_sizes, int n_in,
                              void* d_out, int out_size, void* d_ws, size_t ws_size,
                              hipStream_t stream) {
  const float* target    = (const float*)d_in[0];
  const float* input_    = (const float*)d_in[1];   // feats [512,512]
  const float* aprob     = (const float*)d_in[2];
  const float* A         = (const float*)d_in[3];
  const float* all_glove = (const float*)d_in[4];
  const float* Wword = (const float*)d_in[5];  const float* bword = (const float*)d_in[6];
  const float* Wg    = (const float*)d_in[7];  const float* bg    = (const float*)d_in[8];
  const float* Wa    = (const float*)d_in[9];  const float* ba    = (const float*)d_in[10];
  const float* W0    = (const float*)d_in[11];
  const float* W1    = (const float*)d_in[12];
  const float* W2    = (const float*)d_in[13];
  const float* Wfm   = (const float*)d_in[14]; const float* bfm   = (const float*)d_in[15];
  const float* Wf    = (const float*)d_in[16]; const float* bf    = (const float*)d_in[17];
  const float* Wih   = (const float*)d_in[18]; const float* Whh   = (const float*)d_in[19];
  const float* bih   = (const float*)d_in[20]; const float* bhh   = (const float*)d_in[21];
  const float* Wa1   = (const float*)d_in[22]; const float* ba1   = (const float*)d_in[23];
  const float* Wa2   = (const float*)d_in[24]; const float* ba2   = (const float*)d_in[25];
  const float* Wc1   = (const float*)d_in[26]; const float* bc1   = (const float*)d_in[27];
  const float* Wc2   = (const float*)d_in[28]; const float* bc2   = (const float*)d_in[29];
  float* out = (float*)d_out;

  // ---- workspace carve (floats) ----
  float* p = (float*)d_ws;
  float* glove_e = p; p += BB * GEMB;
  float* act_e   = p; p += BB * ADIM;
  float* wordP   = p; p += NPAD * GEMB;
  float* AwP     = p; p += NPAD * GEMB;
  float* Apad    = p; p += NPAD * NPAD;
  float* rpad    = p; p += NPAD;
  float* G1      = p; p += NPAD * 256;
  float* F1      = p; p += BB * 256;
  float* vbuf    = p; p += BB * NPAD;
  float* x3      = p; p += BB * NPAD;
  float* WfmP    = p; p += GCND * NPAD;
  float* gcn_e   = p; p += BB * GCND;
  float* fusedP  = p; p += BB * FPAD;
  float* WfP     = p; p += HD * FPAD;
  float* outb    = p; p += BB * HD;
  float* bsum    = p; p += 4 * HD;
  float* Xih     = p; p += (long)BB * 4 * HD;
  float* h0      = p; p += BSZ * HD;     // h0,h1,cbuf contiguous (zeroed together)
  float* h1      = p; p += BSZ * HD;
  float* cbuf    = p; p += BSZ * HD;
  float* sf      = p; p += BB * HD;
  float* a1      = p; p += BB * 128;
  float* c1      = p; p += BB * 128;
  float* x1buf   = p; p += (long)CHUNK * NPAD * 256;
  float* y1buf   = p; p += (long)CHUNK * NPAD * 256;

  const int TPB = 256;

  // ---- prep ----
  k_pack_apad<<<cdiv(NPAD * NPAD, TPB), TPB, 0, stream>>>(A, Apad);
  k_rowsum<<<1, NPAD, 0, stream>>>(Apad, rpad);
  k_word<<<cdiv(NPAD * GEMB, TPB), TPB, 0, stream>>>(all_glove, Wword, bword, wordP);
  // Aw = Apad @ wordP
  gemm_xy_b<<<dim3(GEMB / 16, NPAD / 16, 1), 32, 0, stream>>>(
      Apad, wordP, AwP, NPAD, NPAD, GEMB, GEMB, 0, 0);
  // G1 = Aw @ W0[:,512:576]^T   (M=112 -> 1x1 tiles)
  gemm_xwt<<<dim3(256 / 16, NPAD / 16), 32, 0, stream>>>(
      AwP, W0 + RESD, nullptr, G1, GEMB, GEMB, RESD + GEMB, 256, 0);
  // F1 = feats @ W0[:,:512]^T
  gemm_xwt2<<<dim3(256 / 32, BB / 32), 32, 0, stream>>>(
      input_, W0, nullptr, F1, RESD, RESD, RESD + GEMB, 256, 0);
  // glove_e = relu(target @ Wg^T + bg)
  gemm_xwt2<<<dim3(GEMB / 32, BB / 32), 32, 0, stream>>>(
      target, Wg, bg, glove_e, GLOVE, GLOVE, GLOVE, GEMB, 1);
  k_acte<<<cdiv(BB * ADIM, TPB), TPB, 0, stream>>>(aprob, Wa, ba, act_e);

  // ---- GCN middle section, chunked over batch ----
  for (int c = 0; c < BB / CHUNK; ++c) {
    int b0 = c * CHUNK;
    k_x1<<<cdiv(CHUNK * NPAD * 256, TPB), TPB, 0, stream>>>(F1, G1, rpad, x1buf, b0);
    // y1[b] = Apad @ x1[b]
    gemm_xy_b<<<dim3(256 / 16, NPAD / 16, CHUNK), 32, 0, stream>>>(
        Apad, x1buf, y1buf, NPAD, NPAD, 256, 256,
        (long)NPAD * 256, (long)NPAD * 256);
    // x2 = relu(y1 @ W1^T)  (reuse x1buf; M=14336 multiple of 32)
    gemm_xwt2<<<dim3(256 / 32, CHUNK * NPAD / 32), 32, 0, stream>>>(
        y1buf, W1, nullptr, x1buf, 256, 256, 256, 256, 1);
    // v[b,m] = x2[b,m,:] . W2
    k_vdot<<<cdiv(CHUNK * NPAD, TPB), TPB, 0, stream>>>(x1buf, W2, vbuf, b0);
  }
  // x3 = relu(v @ Apad^T)  (N=112 -> 1x1 tiles)
  gemm_xwt<<<dim3(NPAD / 16, BB / 16), 32, 0, stream>>>(
      vbuf, Apad, nullptr, x3, NPAD, NPAD, NPAD, NPAD, 1);
  k_wfmp<<<cdiv(GCND * NPAD, TPB), TPB, 0, stream>>>(Wfm, WfmP);
  // gcn_e = x3 @ WfmP^T + bfm
  gemm_xwt2<<<dim3(GCND / 32, BB / 32), 32, 0, stream>>>(
      x3, WfmP, bfm, gcn_e, NPAD, NPAD, NPAD, GCND, 0);

  // ---- fused layer ----
  k_fused<<<cdiv(BB * FPAD, TPB), TPB, 0, stream>>>(input_, glove_e, act_e, gcn_e, fusedP);
  k_wfp<<<cdiv(HD * FPAD, TPB), TPB, 0, stream>>>(Wf, WfP);
  gemm_xwt2<<<dim3(HD / 32, BB / 32), 32, 0, stream>>>(
      fusedP, WfP, bf, outb, FPAD, FPAD, FPAD, HD, 1);

  // ---- LSTM ----
  k_bsum<<<cdiv(4 * HD, TPB), TPB, 0, stream>>>(bih, bhh, bsum);
  gemm_xwt2<<<dim3(4 * HD / 32, BB / 32), 32, 0, stream>>>(
      outb, Wih, bsum, Xih, HD, HD, HD, 4 * HD, 0);
  k_zero<<<cdiv(3 * BSZ * HD, TPB), TPB, 0, stream>>>(h0, 3 * BSZ * HD);
  float* hping[2] = {h0, h1};
  for (int t = 0; t < TT; ++t) {
    // one fused launch per step: 32 j-slices x 4 gate-waves, LDS exchange
    k_lstm_step<<<32, 128, 0, stream>>>(Whh, Xih, hping[t & 1], hping[(t + 1) & 1],
                                        cbuf, sf, t);
  }
  // after t=31, latest h lives in hping[0] == h0

  // ---- heads ----
  gemm_xwt2<<<dim3(128 / 32, BB / 32), 32, 0, stream>>>(
      sf, Wa1, ba1, a1, HD, HD, HD, 128, 1);
  k_actor<<<cdiv(BB * ASP, TPB), TPB, 0, stream>>>(a1, Wa2, ba2, out);
  gemm_xwt2<<<dim3(128 / 32, BB / 32), 32, 0, stream>>>(
      sf, Wc1, bc1, c1, HD, HD, HD, 128, 1);
  k_critic<<<cdiv(BB, TPB), TPB, 0, stream>>>(c1, Wc2, bc2, out + 3072);
  k_hcout<<<cdiv(BSZ * HD, TPB), TPB, 0, stream>>>(h0, cbuf, out + 3584, out + 11776);
}